// MiniSpinNet_59150289601017
// MI455X (gfx1250) — compile-verified
//
#include <hip/hip_runtime.h>
#include <hip/hip_bf16.h>
#include <math.h>

// ---------------------------------------------------------------------------
// MiniSpinNet forward for MI455X (gfx1250, wave32).
// Design notes:
//  * ~3.3 GFLOP of conv work, <= ~10 MB live data -> fully L2 resident
//    (192 MB L2); HBM roofline irrelevant. Latency-bound, so convs use the
//    full-precision V_WMMA_F32_16X16X4_F32 matrix path (f32 in, f32 acc).
//  * Radius/voxel selection uses wave32 ballot + cross-wave prefix sums.
//  * All conv output row counts are exact multiples of 16 so every wave runs
//    WMMA with EXEC all-ones (required by the ISA); per-lane gathers use
//    clamped addresses multiplied by a 0/1 scale to avoid divergence.
// ---------------------------------------------------------------------------

#define N_PTS_C   20000
#define N_KPTS_C  128
#define NVOX      420        // 3*7*20
#define PATCH_S   512
#define VOX_S     30
#define VOXEL_R2  ((0.8f/3.0f)*(0.8f/3.0f))
#define BN_RG     0.9999950000374997f   // 1/sqrt(1+1e-5)
#define PI_F      3.14159265358979323846f

typedef __attribute__((ext_vector_type(2))) float v2f;
typedef __attribute__((ext_vector_type(8))) float v8f;

// ---------------------------------------------------------------------------
// Kernel 1: per-keypoint patch selection + voxel grouping + PointNet+maxpool.
// One block (256 threads = 8 waves) per keypoint.
// Output x: [128][420][16]  (pos-major, channel contiguous)
// ---------------------------------------------------------------------------
__global__ __launch_bounds__(256) void group_pointnet_kernel(
    const float* __restrict__ pts, const float* __restrict__ kpts,
    const float* __restrict__ pW,  const float* __restrict__ pb,
    const float* __restrict__ pg,  const float* __restrict__ pbb,
    float* __restrict__ xout)
{
  __shared__ float s_dx[PATCH_S * 3];
  __shared__ int   s_sel[PATCH_S];
  __shared__ int   s_wcnt[8];
  __shared__ int   s_count;
  __shared__ float s_w[48], s_b[16], s_g[16], s_bb[16];

  const int m    = blockIdx.x;
  const int tid  = threadIdx.x;
  const int lane = tid & 31;
  const int wid  = tid >> 5;

  if (tid < 48)                 s_w[tid]        = pW[tid];
  if (tid >= 64  && tid < 80)   s_b[tid - 64]   = pb[tid - 64];
  if (tid >= 96  && tid < 112)  s_g[tid - 96]   = pg[tid - 96] * BN_RG;
  if (tid >= 128 && tid < 144)  s_bb[tid - 128] = pbb[tid - 128];
  if (tid == 255)               s_count = 0;

  const float kx = kpts[m*3+0], ky = kpts[m*3+1], kz = kpts[m*3+2];
  __syncthreads();

  // ---- phase 1: first 512 in-radius point indices, in index order ----
  for (int base = 0; base < N_PTS_C; base += 256) {
    const int i = base + tid;
    bool valid = false;
    if (i < N_PTS_C) {
      const float dx = pts[i*3+0] - kx;
      const float dy = pts[i*3+1] - ky;
      const float dz = pts[i*3+2] - kz;
      valid = (dx*dx + dy*dy + dz*dz) < 4.0f;   // DES_R^2, strict <
    }
    const unsigned long long mask = __ballot(valid);
    if (lane == 0) s_wcnt[wid] = __popcll(mask);
    __syncthreads();
    int pos = s_count + (int)__popcll(mask & ((1ull << lane) - 1ull));
    for (int ww = 0; ww < wid; ++ww) pos += s_wcnt[ww];
    if (valid && pos < PATCH_S) s_sel[pos] = i;
    __syncthreads();
    if (tid == 0) {
      int tot = 0;
      for (int ww = 0; ww < 8; ++ww) tot += s_wcnt[ww];
      s_count += tot;
    }
    __syncthreads();
    if (s_count >= PATCH_S) break;              // uniform
  }

  const int nvalid = min(s_count, PATCH_S);
  const int first  = (nvalid > 0) ? s_sel[0] : 0;

  // ---- phase 2: build dx = (patch - kpt)/DES_R with reference masking ----
  // slot 0: first valid (or pts[0]); padding slots and slot 511 -> keypoint -> 0
  for (int s = tid; s < PATCH_S; s += 256) {
    float vx = 0.f, vy = 0.f, vz = 0.f;
    int load = -1;
    if (s == 0)                               load = first;
    else if (s < nvalid && s != PATCH_S - 1)  load = s_sel[s];
    if (load >= 0) {
      vx = (pts[load*3+0] - kx) * 0.5f;
      vy = (pts[load*3+1] - ky) * 0.5f;
      vz = (pts[load*3+2] - kz) * 0.5f;
    }
    s_dx[s*3+0] = vx; s_dx[s*3+1] = vy; s_dx[s*3+2] = vz;
  }
  __syncthreads();

  // ---- phase 3: voxel grouping + rotation + PointNet(3->16) + maxpool ----
  for (int v = tid; v < NVOX; v += 256) {
    const int ri = v / 140;
    const int re = v - ri * 140;
    const int ei = re / 20;
    const int ai = re - ei * 20;
    const float r     = (ri + 0.5f) * (1.0f / 3.0f);
    const float beta  = (ei + 0.5f) * (PI_F / 7.0f);
    const float alpha = ai * (2.0f * PI_F / 20.0f);
    const float sb = sinf(beta),  cb = cosf(beta);
    const float ca = cosf(alpha), sa = sinf(alpha);
    const float cx = r * sb * ca, cy = r * sb * sa, cz = r * cb;

    float mx[16];
#pragma unroll
    for (int o = 0; o < 16; ++o) mx[o] = -3.4e38f;

    int cnt = 0;
    for (int s = 0; s < PATCH_S && cnt < VOX_S; ++s) {
      const float px = s_dx[s*3+0] - cx;
      const float py = s_dx[s*3+1] - cy;
      const float pz = s_dx[s*3+2] - cz;
      if (px*px + py*py + pz*pz <= VOXEL_R2) {
        ++cnt;
        // Rz(-alpha) applied to (point - center)
        const float rx =  ca * px + sa * py;
        const float ry = -sa * px + ca * py;
        const float rz =  pz;
#pragma unroll
        for (int o = 0; o < 16; ++o) {
          float t = s_w[o*3]*rx + s_w[o*3+1]*ry + s_w[o*3+2]*rz + s_b[o];
          t = fmaxf(t * s_g[o] + s_bb[o], 0.f);
          mx[o] = fmaxf(mx[o], t);
        }
      }
    }
    if (cnt < VOX_S) {   // zero-padded slots contribute relu(bn(bias))
#pragma unroll
      for (int o = 0; o < 16; ++o)
        mx[o] = fmaxf(mx[o], fmaxf(s_b[o]*s_g[o] + s_bb[o], 0.f));
    }
    float* xp = xout + ((size_t)m * NVOX + v) * 16;
#pragma unroll
    for (int o = 0; o < 16; ++o) xp[o] = mx[o];
  }
}

// ---------------------------------------------------------------------------
// Kernel 2: implicit-GEMM conv with V_WMMA_F32_16X16X4_F32.
// One wave = one 16-row x 32-output-channel tile.
//   in : [M][DIN*140][CIN]    out : [M][DOUT*140][32]
//   Wt : [32][CIN][KD][3][3]  (KD=1 for 2D convs)
//   rad: zero-pad (c3a) / valid (c3b); ele: zero-pad +-1; azi: circular wrap.
// ---------------------------------------------------------------------------
template <int CIN, int DIN, int DOUT, int KD>
__global__ __launch_bounds__(256) void conv_wmma_kernel(
    const float* __restrict__ in,  const float* __restrict__ Wt,
    const float* __restrict__ bias, const float* __restrict__ gam,
    const float* __restrict__ bet,  float* __restrict__ out)
{
  const int lane = threadIdx.x & 31;
  const int l15  = lane & 15;
  const int hi   = lane >> 4;     // 0 -> K pair (0,1); 1 -> K pair (2,3)
  const int tile = blockIdx.x * (blockDim.x >> 5) + (threadIdx.x >> 5);

  const int rowsPerM = DOUT * 140;
  const int row  = tile * 16 + l15;      // this lane's A-matrix row
  const int m    = row / rowsPerM;
  int rr = row - m * rowsPerM;
  const int dout = rr / 140;  rr -= dout * 140;
  const int h    = rr / 20;
  const int w    = rr - h * 20;

  v8f acc0 = {0.f,0.f,0.f,0.f,0.f,0.f,0.f,0.f};
  v8f acc1 = {0.f,0.f,0.f,0.f,0.f,0.f,0.f,0.f};

  for (int kd = 0; kd < KD; ++kd) {
    int din; bool dok = true;
    if (KD == 3 && DOUT == 3) { din = dout + kd - 1; dok = (din >= 0) && (din < DIN); }
    else if (KD == 3)         { din = kd; }
    else                      { din = 0; }
    const int dc = din < 0 ? 0 : (din > DIN - 1 ? DIN - 1 : din);

    for (int kh = 0; kh < 3; ++kh) {
      const int hin = h + kh - 1;
      const bool hok = (hin >= 0) && (hin < 7);
      const int  hc  = hin < 0 ? 0 : (hin > 6 ? 6 : hin);

      for (int kw = 0; kw < 3; ++kw) {
        int win = w + kw - 1; win = (win + 20) % 20;            // azimuth wrap
        const float scale = (dok && hok) ? 1.0f : 0.0f;          // zero-pad, no branch
        const float* aptr = in + ((size_t)(m * DIN + dc) * 140 + hc * 20 + win) * CIN;

#pragma unroll
        for (int kc = 0; kc < CIN; kc += 4) {
          const int c0 = kc + hi * 2;
          v2f A; A.x = aptr[c0] * scale; A.y = aptr[c0 + 1] * scale;

          const int w0 = ((l15        * CIN + c0) * KD + kd) * 9 + kh * 3 + kw;
          const int w1 = (((16 + l15) * CIN + c0) * KD + kd) * 9 + kh * 3 + kw;
          v2f B0, B1;
          B0.x = Wt[w0]; B0.y = Wt[w0 + KD * 9];
          B1.x = Wt[w1]; B1.y = Wt[w1 + KD * 9];

          acc0 = __builtin_amdgcn_wmma_f32_16x16x4_f32(false, A, false, B0,
                                                       (short)0, acc0, false, false);
          acc1 = __builtin_amdgcn_wmma_f32_16x16x4_f32(false, A, false, B1,
                                                       (short)0, acc1, false, false);
        }
      }
    }
  }

  // epilogue: +bias, BN, relu, store [row][32]
#pragma unroll
  for (int r = 0; r < 8; ++r) {
    const int orow = tile * 16 + (hi ? 8 + r : r);
    int o = l15;
    float v0 = fmaxf((acc0[r] + bias[o]) * (gam[o] * BN_RG) + bet[o], 0.f);
    out[(size_t)orow * 32 + o] = v0;
    o = 16 + l15;
    float v1 = fmaxf((acc1[r] + bias[o]) * (gam[o] * BN_RG) + bet[o], 0.f);
    out[(size_t)orow * 32 + o] = v1;
  }
}

// ---------------------------------------------------------------------------
// Kernel 3: attention head + normalization. One block per keypoint.
//   y: [128][140][32] (post-c2b). out: desc[128][32] then equi[128][32][7][20]
// ---------------------------------------------------------------------------
__global__ __launch_bounds__(256) void head_kernel(
    const float* __restrict__ y,
    const float* __restrict__ p1W, const float* __restrict__ p1b,
    const float* __restrict__ p1g, const float* __restrict__ p1bb,
    const float* __restrict__ p2W, const float* __restrict__ p2b,
    const float* __restrict__ p2g, const float* __restrict__ p2bb,
    float* __restrict__ out)
{
  __shared__ float sW[16 * 32];
  __shared__ float sb[16], sg[16], sbb[16], s2W[16];
  __shared__ float sf[32];
  const int m = blockIdx.x, tid = threadIdx.x;

  for (int i = tid; i < 512; i += 256) sW[i] = p1W[i];
  if (tid < 16) { sb[tid] = p1b[tid]; sg[tid] = p1g[tid] * BN_RG;
                  sbb[tid] = p1bb[tid]; s2W[tid] = p2W[tid]; }
  if (tid < 32) sf[tid] = 0.f;
  __syncthreads();

  if (tid < 140) {
    const float* yp = y + ((size_t)m * 140 + tid) * 32;
    float yv[32];
#pragma unroll
    for (int c = 0; c < 32; ++c) yv[c] = yp[c];

    float acc2 = 0.f;
#pragma unroll
    for (int j = 0; j < 16; ++j) {
      float t = sb[j];
#pragma unroll
      for (int c = 0; c < 32; ++c) t += sW[j*32 + c] * yv[c];
      t = fmaxf(t * sg[j] + sbb[j], 0.f);
      acc2 += s2W[j] * t;
    }
    const float wsc = fmaxf((acc2 + p2b[0]) * (p2g[0] * BN_RG) + p2bb[0], 0.f);

    float nn = 0.f;
#pragma unroll
    for (int c = 0; c < 32; ++c) nn += yv[c] * yv[c];
    const float inv = 1.0f / (sqrtf(nn) + 1e-12f);

    const int hh = tid / 20, ww = tid - hh * 20;
    float* eo = out + N_KPTS_C * 32;
#pragma unroll
    for (int c = 0; c < 32; ++c) {
      atomicAdd(&sf[c], yv[c] * wsc);                       // ds_add_f32
      eo[(((size_t)m * 32 + c) * 7 + hh) * 20 + ww] = yv[c] * inv;
    }
  }
  __syncthreads();

  if (tid < 32) {                      // wave0, all 32 lanes active
    const float f = sf[tid] * (1.0f / 140.0f);
    float sq = f * f;
#pragma unroll
    for (int off = 16; off > 0; off >>= 1) sq += __shfl_xor(sq, off, 32);
    out[m * 32 + tid] = f / (sqrtf(sq) + 1e-12f);
  }
}

// ---------------------------------------------------------------------------
// Host launcher.
// d_in order: pts, kpts, pnt_W,b,g,bb, c3a_W..bb, c3b_W..bb, c2a_W..bb,
//             c2b_W..bb, p1_W..bb, p2_W..bb
// Workspace aliasing (floats): x[0,860160) y1[860160,2580480)
//   y2 -> 0 (x dead), y3a -> 860160 (y1 dead), y3b -> 0 (y2 dead).
// ---------------------------------------------------------------------------
extern "C" void kernel_launch(void* const* d_in, const int* in_sizes, int n_in,
                              void* d_out, int out_size, void* d_ws, size_t ws_size,
                              hipStream_t stream)
{
  (void)in_sizes; (void)n_in; (void)out_size; (void)ws_size;
  const float* pts   = (const float*)d_in[0];
  const float* kpts  = (const float*)d_in[1];
  const float* pntW  = (const float*)d_in[2];
  const float* pntB  = (const float*)d_in[3];
  const float* pntG  = (const float*)d_in[4];
  const float* pntBB = (const float*)d_in[5];
  const float* c3aW  = (const float*)d_in[6];
  const float* c3aB  = (const float*)d_in[7];
  const float* c3aG  = (const float*)d_in[8];
  const float* c3aBB = (const float*)d_in[9];
  const float* c3bW  = (const float*)d_in[10];
  const float* c3bB  = (const float*)d_in[11];
  const float* c3bG  = (const float*)d_in[12];
  const float* c3bBB = (const float*)d_in[13];
  const float* c2aW  = (const float*)d_in[14];
  const float* c2aB  = (const float*)d_in[15];
  const float* c2aG  = (const float*)d_in[16];
  const float* c2aBB = (const float*)d_in[17];
  const float* c2bW  = (const float*)d_in[18];
  const float* c2bB  = (const float*)d_in[19];
  const float* c2bG  = (const float*)d_in[20];
  const float* c2bBB = (const float*)d_in[21];
  const float* p1W   = (const float*)d_in[22];
  const float* p1B   = (const float*)d_in[23];
  const float* p1G   = (const float*)d_in[24];
  const float* p1BB  = (const float*)d_in[25];
  const float* p2W   = (const float*)d_in[26];
  const float* p2B   = (const float*)d_in[27];
  const float* p2G   = (const float*)d_in[28];
  const float* p2BB  = (const float*)d_in[29];

  float* ws  = (float*)d_ws;
  float* x   = ws;                 // 128*420*16 = 860160 f
  float* y1  = ws + 860160;        // 128*420*32 = 1720320 f
  float* y2  = ws;                 // 128*140*32 =  573440 f (x dead)
  float* y3a = ws + 860160;        // reuse y1 region
  float* y3b = ws;                 // reuse y2 region
  float* outp = (float*)d_out;

  group_pointnet_kernel<<<128, 256, 0, stream>>>(pts, kpts, pntW, pntB, pntG, pntBB, x);

  // c3a: 16ch, 3x3x3, rad zero-pad -> rows 128*420 = 3360 tiles / 8 waves
  conv_wmma_kernel<16, 3, 3, 3><<<420, 256, 0, stream>>>(x,  c3aW, c3aB, c3aG, c3aBB, y1);
  // c3b: 32ch, 3x3x3, rad valid   -> rows 128*140 = 1120 tiles
  conv_wmma_kernel<32, 3, 1, 3><<<140, 256, 0, stream>>>(y1, c3bW, c3bB, c3bG, c3bBB, y2);
  // c2a / c2b: 32ch, 3x3
  conv_wmma_kernel<32, 1, 1, 1><<<140, 256, 0, stream>>>(y2,  c2aW, c2aB, c2aG, c2aBB, y3a);
  conv_wmma_kernel<32, 1, 1, 1><<<140, 256, 0, stream>>>(y3a, c2bW, c2bB, c2bG, c2bBB, y3b);

  head_kernel<<<128, 256, 0, stream>>>(y3b, p1W, p1B, p1G, p1BB, p2W, p2B, p2G, p2BB, outp);
}